// GINLayer_80221399155534
// MI455X (gfx1250) — compile-verified
//
#include <hip/hip_runtime.h>

typedef __attribute__((ext_vector_type(2))) float v2f;
typedef __attribute__((ext_vector_type(8))) float v8f;

#define D_FEAT 128
#define HST 132  // LDS row stride (floats), padded to avoid bank conflicts

// ---------------------------------------------------------------------------
// Kernel 1: agg = X  (float4 copy; initializes the scatter accumulator)
// ---------------------------------------------------------------------------
__global__ __launch_bounds__(256) void init_agg_kernel(const float4* __restrict__ src,
                                                       float4* __restrict__ dst, int n4) {
  int i = blockIdx.x * blockDim.x + threadIdx.x;
  if (i < n4) dst[i] = src[i];
}

// ---------------------------------------------------------------------------
// Kernel 2: undirected edge scatter-add.
// One wave (32 lanes) per edge; lane handles 4 consecutive features.
// agg[a] += X[b];  agg[b] += X[a];  via global_atomic_add_f32 (hits in L2:
// X and agg together are ~102 MB, well inside the 192 MB L2).
// ---------------------------------------------------------------------------
__global__ __launch_bounds__(256) void edge_scatter_kernel(const float* __restrict__ X,
                                                           const int* __restrict__ ra,
                                                           const int* __restrict__ rb,
                                                           float* __restrict__ agg, int nE) {
  int t = blockIdx.x * blockDim.x + threadIdx.x;
  int e = t >> 5;
  int lane = t & 31;
  if (e >= nE) return;
  int a = ra[e];
  int b = rb[e];
  int f = lane << 2;
  const float4 xb = *(const float4*)(X + (size_t)b * D_FEAT + f);
  const float4 xa = *(const float4*)(X + (size_t)a * D_FEAT + f);
  float* pa = agg + (size_t)a * D_FEAT + f;
  float* pb = agg + (size_t)b * D_FEAT + f;
  atomicAdd(pa + 0, xb.x);
  atomicAdd(pa + 1, xb.y);
  atomicAdd(pa + 2, xb.z);
  atomicAdd(pa + 3, xb.w);
  atomicAdd(pb + 0, xa.x);
  atomicAdd(pb + 1, xa.y);
  atomicAdd(pb + 2, xa.z);
  atomicAdd(pb + 3, xa.w);
}

// ---------------------------------------------------------------------------
// Kernel 3: fused MLP via V_WMMA_F32_16X16X4_F32 (fp32 matrix core path).
// Workgroup = 256 threads = 8 waves; block owns 16 rows (M-tile).
// Each wave owns one 16-wide N-tile (8 tiles = 128 cols).
// GEMM1 -> bias+relu -> LDS -> GEMM2 -> bias -> out.
//
// f32 WMMA operand layout (ISA 7.12.2):
//   A 16x4 : lanes 0-15 row M=lane, VGPR{0,1} = K{0,1}; lanes 16-31: K{2,3}
//   B 4x16 : lanes 0-15 col N=lane, VGPR{0,1} = K{0,1}; lanes 16-31: K{2,3}
//   C/D    : lanes 0-15 N=lane, VGPR r = row r; lanes 16-31: row r+8
// ---------------------------------------------------------------------------
__global__ __launch_bounds__(256) void gin_mlp_wmma_kernel(
    const float* __restrict__ agg, const float* __restrict__ Wh,
    const float* __restrict__ bh, const float* __restrict__ Wo,
    const float* __restrict__ bo, float* __restrict__ out, int nNodes) {
  __shared__ float As[16 * HST];
  __shared__ float Hs[16 * HST];

  const int tid = threadIdx.x;
  const int lane = tid & 31;
  const int wave = tid >> 5;       // 0..7 -> N-tile
  const int half = lane >> 4;      // K sub-block selector
  const int l16 = lane & 15;
  const int rowBase = blockIdx.x * 16;

  // Stage A tile (16 x 128) into LDS, fully coalesced float4 loads.
  {
    int fi0 = tid * 4;
    #pragma unroll
    for (int it = 0; it < 2; ++it) {
      int fi = fi0 + it * 1024;          // 2048 floats total
      int r = fi >> 7;
      int c = fi & 127;
      int row = rowBase + r;
      if (row >= nNodes) row = nNodes - 1;  // uniform-safe clamp (N%16==0 anyway)
      float4 v = *(const float4*)(agg + (size_t)row * D_FEAT + c);
      *(float4*)(&As[r * HST + c]) = v;     // (r*132+c)*4 is 16B aligned
    }
  }
  __syncthreads();

  const int n = wave * 16 + l16;           // this lane's output column

  // ---- GEMM1: H = relu(A @ Wh + bh) ----
  v8f c1 = {0.f, 0.f, 0.f, 0.f, 0.f, 0.f, 0.f, 0.f};
  #pragma unroll
  for (int k = 0; k < D_FEAT; k += 4) {
    const int ka = k + half * 2;
    v2f a, b;
    a.x = As[l16 * HST + ka];
    a.y = As[l16 * HST + ka + 1];
    b.x = Wh[(size_t)ka * D_FEAT + n];
    b.y = Wh[(size_t)(ka + 1) * D_FEAT + n];
    c1 = __builtin_amdgcn_wmma_f32_16x16x4_f32(false, a, false, b,
                                               (short)0, c1, false, false);
  }
  const float bhv = bh[n];
  #pragma unroll
  for (int r = 0; r < 8; ++r) {
    float v = c1[r] + bhv;
    v = v > 0.f ? v : 0.f;                 // relu
    const int mm = r + half * 8;
    Hs[mm * HST + n] = v;
  }
  __syncthreads();

  // ---- GEMM2: out = H @ Wo + bo ----
  const float bov = bo[n];
  v8f c2 = {bov, bov, bov, bov, bov, bov, bov, bov};
  #pragma unroll
  for (int k = 0; k < D_FEAT; k += 4) {
    const int ka = k + half * 2;
    v2f a, b;
    a.x = Hs[l16 * HST + ka];
    a.y = Hs[l16 * HST + ka + 1];
    b.x = Wo[(size_t)ka * D_FEAT + n];
    b.y = Wo[(size_t)(ka + 1) * D_FEAT + n];
    c2 = __builtin_amdgcn_wmma_f32_16x16x4_f32(false, a, false, b,
                                               (short)0, c2, false, false);
  }
  #pragma unroll
  for (int r = 0; r < 8; ++r) {
    const int mm = r + half * 8;
    const int row = rowBase + mm;
    if (row < nNodes) out[(size_t)row * D_FEAT + n] = c2[r];
  }
}

// ---------------------------------------------------------------------------
// Host-side launch
// ---------------------------------------------------------------------------
extern "C" void kernel_launch(void* const* d_in, const int* in_sizes, int n_in,
                              void* d_out, int out_size, void* d_ws, size_t ws_size,
                              hipStream_t stream) {
  const float* X  = (const float*)d_in[0];   // [N, 128]
  const int*   ra = (const int*)d_in[1];     // [E]
  const int*   rb = (const int*)d_in[2];     // [E]
  const float* Wh = (const float*)d_in[3];   // [128, 128]
  const float* bh = (const float*)d_in[4];   // [128]
  const float* Wo = (const float*)d_in[5];   // [128, 128]
  const float* bo = (const float*)d_in[6];   // [128]
  float* out = (float*)d_out;                // [N, 128]
  float* agg = (float*)d_ws;                 // [N, 128] scratch accumulator

  const int nNodes = in_sizes[0] / D_FEAT;   // 100000
  const int nEdges = in_sizes[1];            // 600000

  // 1) agg = X
  const int n4 = nNodes * (D_FEAT / 4);
  init_agg_kernel<<<(n4 + 255) / 256, 256, 0, stream>>>(
      (const float4*)X, (float4*)agg, n4);

  // 2) scatter-add both edge directions (wave per edge)
  const long long nThreads = (long long)nEdges * 32;
  edge_scatter_kernel<<<(int)((nThreads + 255) / 256), 256, 0, stream>>>(
      X, ra, rb, agg, nEdges);

  // 3) fused WMMA MLP
  gin_mlp_wmma_kernel<<<(nNodes + 15) / 16, 256, 0, stream>>>(
      agg, Wh, bh, Wo, bo, out, nNodes);
}